// LIFSpike_18459769438561
// MI455X (gfx1250) — compile-verified
//
#include <hip/hip_runtime.h>

// LIF recurrence over the last (time) axis.
//   u_new = TAU*u*(1-o) + x_t ; spike = (u_new > VTH)
// x layout: [..., STEPS=4] with steps contiguous -> one float4 per neuron.
// Memory bound: ~268 MB one-pass stream -> ~11.5 us at 23.3 TB/s.

#define LIF_TAU 0.2f
#define LIF_VTH 0.5f

typedef float v4f __attribute__((ext_vector_type(4)));
typedef int   v4i __attribute__((ext_vector_type(4)));

// ---- CDNA5 async global->LDS path (gfx1250 only), with safe fallback ------
#if defined(__gfx1250__) && __has_builtin(__builtin_amdgcn_global_load_async_to_lds_b128)
#define LIF_USE_ASYNC_LDS 1
#else
#define LIF_USE_ASYNC_LDS 0
#endif

__global__ __launch_bounds__(256) void lif_spike_kernel(
    const v4f* __restrict__ x, v4f* __restrict__ out, int n4) {
  const int tid = threadIdx.x;
  const int i = blockIdx.x * 256 + tid;
  if (i >= n4) return;

  v4f xt;

#if LIF_USE_ASYNC_LDS
  __shared__ v4f tile[256];
  {
    // Stage this wave's 16B/lane tile through LDS via the gfx1250 async
    // data mover (GLOBAL_LOAD_ASYNC_TO_LDS_B128, tracked by ASYNCcnt).
    // Builtin signature (per compiler diagnostic): arg0 = generic v4i*,
    // arg1 = LDS-side pointer, then imm offset + imm cpol.
    v4i* gsrc = (v4i*)(x + i);
    auto ldst = (__attribute__((address_space(3))) v4i*)(&tile[tid]);
    __builtin_amdgcn_global_load_async_to_lds_b128(gsrc, ldst, 0, 0);
#if __has_builtin(__builtin_amdgcn_s_wait_asynccnt)
    __builtin_amdgcn_s_wait_asynccnt(0);
#else
    asm volatile("s_wait_asynccnt 0" ::: "memory");
#endif
    xt = tile[tid];  // ds_load_b128 (compiler inserts s_wait_dscnt before use)
  }
#else
  // Fallback: direct non-temporal 128-bit load (TH=NT: no L2 pollution,
  // data is touched exactly once).
  xt = __builtin_nontemporal_load(x + i);
#endif

  // Unrolled 4-step LIF recurrence; carry starts at (u=0, o=0).
  float u = xt.x;                         // TAU*0*(1-0) + x0
  float s0 = (u > LIF_VTH) ? 1.0f : 0.0f;
  u = LIF_TAU * u * (1.0f - s0) + xt.y;
  float s1 = (u > LIF_VTH) ? 1.0f : 0.0f;
  u = LIF_TAU * u * (1.0f - s1) + xt.z;
  float s2 = (u > LIF_VTH) ? 1.0f : 0.0f;
  u = LIF_TAU * u * (1.0f - s2) + xt.w;
  float s3 = (u > LIF_VTH) ? 1.0f : 0.0f;

  v4f o;
  o.x = s0; o.y = s1; o.z = s2; o.w = s3;

  // Non-temporal B128 store: spikes are never re-read by us.
  __builtin_nontemporal_store(o, out + i);
}

extern "C" void kernel_launch(void* const* d_in, const int* in_sizes, int n_in,
                              void* d_out, int out_size, void* d_ws, size_t ws_size,
                              hipStream_t stream) {
  (void)n_in; (void)out_size; (void)d_ws; (void)ws_size;
  const float* x = (const float*)d_in[0];   // [64,128,32,32,4] f32
  // d_in[1] ("k") is unused by the reference.
  float* out = (float*)d_out;

  const int n4 = in_sizes[0] / 4;           // neurons (float4 tiles): 8,388,608
  const int block = 256;                    // 8 wave32s per workgroup
  const int grid = (n4 + block - 1) / block;

  lif_spike_kernel<<<grid, block, 0, stream>>>(
      (const v4f*)x, (v4f*)out, n4);
}